// OuterProductMean_33913061769436
// MI455X (gfx1250) — compile-verified
//
#include <hip/hip_runtime.h>
#include <hip/hip_bf16.h>

#define S     512
#define NRES  384
#define CIN   256
#define CO    32
#define FOUT  128

typedef __attribute__((ext_vector_type(2))) float v2f;
typedef __attribute__((ext_vector_type(4))) float f32x4;
typedef __attribute__((ext_vector_type(8))) float v8f;

static __device__ __forceinline__ v8f wmma_f32(v2f a, v2f b, v8f c) {
    // V_WMMA_F32_16X16X4_F32 : D = A(16x4) * B(4x16) + C(16x16)
    return __builtin_amdgcn_wmma_f32_16x16x4_f32(false, a, false, b, (short)0, c, false, false);
}

// ---------------------------------------------------------------------------
// Async global->LDS (CDNA5 GLOBAL_LOAD_ASYNC_TO_LDS_B128, ASYNCcnt-tracked).
// Probe-derived signature: (int4 AS1* global_src, int4 AS3* lds_dst,
//                           imm offset, imm cpol)
// ---------------------------------------------------------------------------
#if __has_builtin(__builtin_amdgcn_global_load_async_to_lds_b128) && \
    __has_builtin(__builtin_amdgcn_s_wait_asynccnt)
#define HAVE_ASYNC_LDS 1
#else
#define HAVE_ASYNC_LDS 0
#endif

#if HAVE_ASYNC_LDS
typedef int i32x4g __attribute__((vector_size(16)));
static __device__ __forceinline__ void async_b128(const float* g, const float* l) {
    __builtin_amdgcn_global_load_async_to_lds_b128(
        (__attribute__((address_space(1))) i32x4g*)(uintptr_t)g,
        (__attribute__((address_space(3))) i32x4g*)(uint32_t)(uintptr_t)l,
        0, 0);
}
#endif

// ---------------------------------------------------------------------------
// Kernel 1: LayerNorm + left/right projection (+mask), transpose-store to
// L/R buffers in [n][c][s] layout (s contiguous) for the main GEMM kernel.
// ---------------------------------------------------------------------------
__global__ __launch_bounds__(256)
void ln_proj_kernel(const float* __restrict__ act, const float* __restrict__ mask,
                    const float* __restrict__ ln_scale, const float* __restrict__ ln_bias,
                    const float* __restrict__ left_w, const float* __restrict__ right_w,
                    float* __restrict__ Lbuf, float* __restrict__ Rbuf)
{
    __shared__ float shn[8][CIN];
    __shared__ float shtL[8][CO];
    __shared__ float shtR[8][CO];

    const int n    = blockIdx.x >> 6;      // grid = 384 * 64
    const int sb   = blockIdx.x & 63;
    const int wave = threadIdx.x >> 5;
    const int lane = threadIdx.x & 31;
    const int s    = sb * 8 + wave;

    const float* row = act + ((size_t)s * NRES + n) * CIN;

    float x[8];
    float sum = 0.f, sumsq = 0.f;
#pragma unroll
    for (int j = 0; j < 8; ++j) {
        x[j] = row[lane + 32 * j];
        sum += x[j];
        sumsq += x[j] * x[j];
    }
#pragma unroll
    for (int off = 16; off >= 1; off >>= 1) {
        sum   += __shfl_xor(sum, off, 32);
        sumsq += __shfl_xor(sumsq, off, 32);
    }
    const float mu   = sum * (1.0f / CIN);
    const float var  = sumsq * (1.0f / CIN) - mu * mu;
    const float rstd = rsqrtf(var + 1e-5f);

#pragma unroll
    for (int j = 0; j < 8; ++j) {
        const int c = lane + 32 * j;
        shn[wave][c] = (x[j] - mu) * rstd * ln_scale[c] + ln_bias[c];
    }
    __syncthreads();

    const float m = mask[(size_t)s * NRES + n];
    const float* wl = left_w  + lane * CIN;
    const float* wr = right_w + lane * CIN;
    float accL = 0.f, accR = 0.f;
#pragma unroll 8
    for (int c = 0; c < CIN; ++c) {
        const float v = shn[wave][c];   // broadcast read
        accL += v * wl[c];
        accR += v * wr[c];
    }
    shtL[wave][lane] = accL * m;
    shtR[wave][lane] = accR * m;
    __syncthreads();

    // cooperative transposed store: 32 c x 8 s per buffer
    const int c  = threadIdx.x >> 3;
    const int sw = threadIdx.x & 7;
    const size_t obase = ((size_t)n * CO + c) * S + sb * 8 + sw;
    Lbuf[obase] = shtL[sw][c];
    Rbuf[obase] = shtR[sw][c];
}

// ---------------------------------------------------------------------------
// Kernel 2: mask normalization, bf16-rounded as in reference:
// recip[b,d] = 1 / (1e-3 + sum_s bf16(mask[s,b]) * bf16(mask[s,d]))
// ---------------------------------------------------------------------------
__global__ __launch_bounds__(256)
void norm_kernel(const float* __restrict__ mask, float* __restrict__ recip)
{
    const int idx = blockIdx.x * 256 + threadIdx.x;
    const int b = idx / NRES;
    const int d = idx % NRES;
    float acc = 0.f;
    for (int s = 0; s < S; ++s) {
        const float mb = __bfloat162float(__float2bfloat16(mask[(size_t)s * NRES + b]));
        const float md = __bfloat162float(__float2bfloat16(mask[(size_t)s * NRES + d]));
        acc += mb * md;
    }
    recip[idx] = 1.0f / (1e-3f + acc);
}

// ---------------------------------------------------------------------------
// Kernel 3 (main): fused outer-product GEMM + output contraction.
// Grid (96,96): 4(b) x 4(d) pairs per WG, 16 waves, one pair per wave.
// Stage 1: outer[c,e] = sum_s L[s,b,c] R[s,d,e] (M=32,N=32,K=512) via 2x2
//          V_WMMA_F32_16X16X4_F32 tiles; L/R chunks double-buffered in LDS
//          via GLOBAL_LOAD_ASYNC_TO_LDS_B128 (ASYNCcnt pipeline).
// Stage 2: out[pair,f] = outer_flat[pair,1024] x W[1024,128] via WMMA,
//          K split across wave halves + LDS reduction, fused bias/norm.
// ---------------------------------------------------------------------------
#define SCHUNK 16
#define NCHUNK (S / SCHUNK)          // 32
#define LPAD   20                    // padded row stride: 16B-aligned, no conflicts
#define BUFSZ  (4 * CO * LPAD)       // 2560 floats (one L or R slab)
#define SLOT   (2 * BUFSZ)           // L+R per double-buffer slot = 5120 floats

__global__ __launch_bounds__(512)
void opm_main_kernel(const float* __restrict__ Lbuf, const float* __restrict__ Rbuf,
                     const float* __restrict__ Wout, const float* __restrict__ biasv,
                     const float* __restrict__ recip, float* __restrict__ out)
{
    __shared__ float smem[16384];    // 64 KB; stage1 uses 2*SLOT=10240, stage2 all
    float* pairsh = smem;            // stage 2: [1024][16]

    const int tid   = threadIdx.x;
    const int wave  = tid >> 5;
    const int lane  = tid & 31;
    const int half  = lane >> 4;     // K sub-half within fragment
    const int l16   = lane & 15;     // M / N index within fragment
    const int bbase = blockIdx.x * 4;
    const int dbase = blockIdx.y * 4;
    const int bi    = wave >> 2;     // which b of the 4
    const int di    = wave & 3;      // which d of the 4

    v8f acc00 = {0.f,0.f,0.f,0.f,0.f,0.f,0.f,0.f};
    v8f acc01 = acc00, acc10 = acc00, acc11 = acc00;

    // per-thread coords for cooperative LDS fill: 2048 floats/slab, 4/thread
    const int i0 = tid * 4;
    const int lb = i0 >> 9;           // 0..3 (b or d index in tile)
    const int lc = (i0 >> 4) & 31;    // channel
    const int ls = i0 & 15;           // s within chunk (multiple of 4)
    const float* gLc = Lbuf + (((size_t)(bbase + lb) * CO + lc) << 9) + ls;
    const float* gRc = Rbuf + (((size_t)(dbase + lb) * CO + lc) << 9) + ls;
    const int ldst = (lb * CO + lc) * LPAD + ls;

    const int arow0 = (bi * CO + l16) * LPAD;
    const int arow1 = (bi * CO + 16 + l16) * LPAD;
    const int brow0 = (di * CO + l16) * LPAD;
    const int brow1 = (di * CO + 16 + l16) * LPAD;

#if HAVE_ASYNC_LDS
    // prologue: chunk 0 -> slot 0
    async_b128(gLc, smem + ldst);
    async_b128(gRc, smem + BUFSZ + ldst);
    for (int i = 0; i < NCHUNK; ++i) {
        if (i + 1 < NCHUNK) {
            const float* gl = gLc + (i + 1) * SCHUNK;
            const float* gr = gRc + (i + 1) * SCHUNK;
            float* slot = smem + ((i + 1) & 1) * SLOT;
            async_b128(gl, slot + ldst);
            async_b128(gr, slot + BUFSZ + ldst);
            __builtin_amdgcn_s_wait_asynccnt(2);   // chunk i landed
        } else {
            __builtin_amdgcn_s_wait_asynccnt(0);
        }
        __syncthreads();
        const float* Ls = smem + (i & 1) * SLOT;
        const float* Rs = Ls + BUFSZ;
#pragma unroll
        for (int k = 0; k < SCHUNK; k += 4) {
            const int koff = k + 2 * half;
            const v2f a0 = *(const v2f*)(Ls + arow0 + koff);
            const v2f a1 = *(const v2f*)(Ls + arow1 + koff);
            const v2f b0 = *(const v2f*)(Rs + brow0 + koff);
            const v2f b1 = *(const v2f*)(Rs + brow1 + koff);
            acc00 = wmma_f32(a0, b0, acc00);
            acc01 = wmma_f32(a0, b1, acc01);
            acc10 = wmma_f32(a1, b0, acc10);
            acc11 = wmma_f32(a1, b1, acc11);
        }
        __syncthreads();
    }
#else
    // fallback: VGPR-staged single-buffer path
    for (int i = 0; i < NCHUNK; ++i) {
        const int s0 = i * SCHUNK;
        const f32x4 u = *(const f32x4*)(gLc + s0);
        const f32x4 w = *(const f32x4*)(gRc + s0);
        *(f32x4*)(smem + ldst) = u;
        *(f32x4*)(smem + BUFSZ + ldst) = w;
        __syncthreads();
        const float* Ls = smem;
        const float* Rs = smem + BUFSZ;
#pragma unroll
        for (int k = 0; k < SCHUNK; k += 4) {
            const int koff = k + 2 * half;
            const v2f a0 = *(const v2f*)(Ls + arow0 + koff);
            const v2f a1 = *(const v2f*)(Ls + arow1 + koff);
            const v2f b0 = *(const v2f*)(Rs + brow0 + koff);
            const v2f b1 = *(const v2f*)(Rs + brow1 + koff);
            acc00 = wmma_f32(a0, b0, acc00);
            acc01 = wmma_f32(a0, b1, acc01);
            acc10 = wmma_f32(a1, b0, acc10);
            acc11 = wmma_f32(a1, b1, acc11);
        }
        __syncthreads();
    }
#endif

    // ---- stage 2: scatter outer tiles into pairsh[ce][pair] ----
#pragma unroll
    for (int r = 0; r < 8; ++r) {
        const int cb = r + 8 * half;          // c within 16-tile
        const int e0 = l16, e1 = 16 + l16;
        pairsh[((cb)      * CO + e0) * 16 + wave] = acc00[r];
        pairsh[((cb)      * CO + e1) * 16 + wave] = acc01[r];
        pairsh[((16 + cb) * CO + e0) * 16 + wave] = acc10[r];
        pairsh[((16 + cb) * CO + e1) * 16 + wave] = acc11[r];
    }
    __syncthreads();

    const int ftile = wave & 7;    // which 16-wide f tile
    const int khalf = wave >> 3;   // K half: 0 or 1
    const int f = ftile * 16 + l16;
    v8f acc2 = {0.f,0.f,0.f,0.f,0.f,0.f,0.f,0.f};

    const int kbeg = khalf * 512;
#pragma unroll 4
    for (int k = kbeg; k < kbeg + 512; k += 4) {
        const int kk = k + 2 * half;
        v2f a, b;
        a.x = pairsh[kk * 16 + l16];          // A[M=pair, K]
        a.y = pairsh[(kk + 1) * 16 + l16];
        b.x = Wout[kk * FOUT + f];            // B[K, N=f]
        b.y = Wout[(kk + 1) * FOUT + f];
        acc2 = wmma_f32(a, b, acc2);
    }
    __syncthreads();

    // ---- reduce the two K-halves via LDS, then epilogue ----
    float* redsh = smem;                      // [8][16][16]
    if (khalf == 1) {
#pragma unroll
        for (int r = 0; r < 8; ++r)
            redsh[(ftile * 16 + r + 8 * half) * 16 + l16] = acc2[r];
    }
    __syncthreads();
    if (khalf == 0) {
        const float bv = biasv[f];
#pragma unroll
        for (int r = 0; r < 8; ++r) {
            const int pair = r + 8 * half;    // M index = pair
            const int b_ = bbase + (pair >> 2);
            const int d_ = dbase + (pair & 3);
            const float val = acc2[r] + redsh[(ftile * 16 + pair) * 16 + l16];
            out[(((size_t)b_ * NRES) + d_) * FOUT + f] =
                (val + bv) * recip[b_ * NRES + d_];
        }
    }
}

// ---------------------------------------------------------------------------
extern "C" void kernel_launch(void* const* d_in, const int* in_sizes, int n_in,
                              void* d_out, int out_size, void* d_ws, size_t ws_size,
                              hipStream_t stream) {
    const float* act      = (const float*)d_in[0];
    const float* mask     = (const float*)d_in[1];
    const float* ln_scale = (const float*)d_in[2];
    const float* ln_bias  = (const float*)d_in[3];
    const float* left_w   = (const float*)d_in[4];
    const float* right_w  = (const float*)d_in[5];
    const float* output_w = (const float*)d_in[6];
    const float* output_b = (const float*)d_in[7];
    float* out = (float*)d_out;

    float* Lbuf  = (float*)d_ws;                         // [384][32][512] f32
    float* Rbuf  = Lbuf + (size_t)NRES * CO * S;         // [384][32][512] f32
    float* recip = Rbuf + (size_t)NRES * CO * S;         // [384][384] f32

    ln_proj_kernel<<<dim3(NRES * (S / 8)), 256, 0, stream>>>(
        act, mask, ln_scale, ln_bias, left_w, right_w, Lbuf, Rbuf);
    norm_kernel<<<dim3((NRES * NRES) / 256), 256, 0, stream>>>(mask, recip);
    opm_main_kernel<<<dim3(NRES / 4, NRES / 4), 512, 0, stream>>>(
        Lbuf, Rbuf, output_w, output_b, recip, out);
}